// DirectVoxGO_1726576856424
// MI455X (gfx1250) — compile-verified
//
#include <hip/hip_runtime.h>
#include <hip/hip_bf16.h>
#include <math.h>

// ---------------- constants from the reference ----------------
#define RES    160
#define RES2   25600            // 160*160
#define RES3   4096000          // 160^3
#define WPB    8                // waves per block (wave32)
#define BLOCK  (WPB * 32)
#define ACT_SHIFT_F (-4.5951198501345898f)   // log(1/(1-0.01) - 1)
// INTERVAL = 0.5, BKGD = 1.0 folded inline below

// Async global->LDS path (CDNA5): guarded so we compile either way.
#if defined(__gfx1250__) && __has_builtin(__builtin_amdgcn_global_load_async_to_lds_b32)
#define HAVE_ASYNC_LDS 1
#else
#define HAVE_ASYNC_LDS 0
#endif

__device__ __forceinline__ float softplus_f(float x) {
    // log1p(exp(x)) in a numerically stable form (matches jax.nn.softplus)
    return fmaxf(x, 0.0f) + log1pf(expf(-fabsf(x)));
}

// ---------------- kernel 1: segment bounds via binary search ----------------
__global__ void bounds_kernel(const int* __restrict__ ray_id, int M, int n_rays,
                              int* __restrict__ bounds) {
    int r = blockIdx.x * blockDim.x + threadIdx.x;
    if (r > n_rays) return;
    // first index i with ray_id[i] >= r   (ray_id is sorted ascending)
    int lo = 0, hi = M;
    while (lo < hi) {
        int mid = (lo + hi) >> 1;
        if (ray_id[mid] < r) lo = mid + 1; else hi = mid;
    }
    bounds[r] = lo;
}

#if HAVE_ASYNC_LDS
typedef __attribute__((address_space(1))) int* gint_p;   // global
typedef __attribute__((address_space(3))) int* lint_p;   // LDS

__device__ __forceinline__ void stage3(const float* __restrict__ xyz, int p, float* slot) {
    const float* g = xyz + 3 * (size_t)p;
    __builtin_amdgcn_global_load_async_to_lds_b32(
        (gint_p)(g + 0), (lint_p)(slot + 0), 0, 0);
    __builtin_amdgcn_global_load_async_to_lds_b32(
        (gint_p)(g + 1), (lint_p)(slot + 1), 0, 0);
    __builtin_amdgcn_global_load_async_to_lds_b32(
        (gint_p)(g + 2), (lint_p)(slot + 2), 0, 0);
}
__device__ __forceinline__ void wait_async3() {
    asm volatile("s_wait_asynccnt 0x3" ::: "memory");
}
__device__ __forceinline__ void wait_async0() {
    asm volatile("s_wait_asynccnt 0x0" ::: "memory");
}
#endif

// ---------------- kernel 2: one wave32 per ray ----------------
__global__ __launch_bounds__(BLOCK)
void render_kernel(const float* __restrict__ xyz,
                   const float* __restrict__ dgrid,
                   const float* __restrict__ kgrid,
                   const int*   __restrict__ bounds,
                   float*       __restrict__ out,
                   int n_rays)
{
    const int lane = threadIdx.x & 31;
    const int wave = threadIdx.x >> 5;
    const int ray  = blockIdx.x * WPB + wave;
    if (ray >= n_rays) return;

    const int start = bounds[ray];
    const int end   = bounds[ray + 1];
    const int nch   = (end > start) ? ((end - start + 31) >> 5) : 0;

#if HAVE_ASYNC_LDS
    __shared__ float xbuf[WPB][2][32 * 3];
    if (nch > 0) {
        int p0 = min(start + lane, end - 1);
        stage3(xyz, p0, &xbuf[wave][0][lane * 3]);   // prime the pipeline
    }
#endif

    float cum = 0.0f;                 // running sum of log(1-alpha) over the ray
    float a0 = 0.0f, a1 = 0.0f, a2 = 0.0f;

    for (int t = 0; t < nch; ++t) {
        const int  i     = start + t * 32 + lane;
        const bool valid = (i < end);

        float px, py, pz;
#if HAVE_ASYNC_LDS
        const int buf = t & 1;
        if (t + 1 < nch) {
            int pn = min(start + (t + 1) * 32 + lane, end - 1);
            stage3(xyz, pn, &xbuf[wave][buf ^ 1][lane * 3]);
            wait_async3();            // 3 newest (next chunk) may remain outstanding
        } else {
            wait_async0();
        }
        px = xbuf[wave][buf][lane * 3 + 0];
        py = xbuf[wave][buf][lane * 3 + 1];
        pz = xbuf[wave][buf][lane * 3 + 2];
#else
        {
            int p = min(i, end - 1);
            px = xyz[3 * (size_t)p + 0];
            py = xyz[3 * (size_t)p + 1];
            pz = xyz[3 * (size_t)p + 2];
            if (t + 1 < nch) {
                int pn = min(i + 32, end - 1);
                __builtin_prefetch(xyz + 3 * (size_t)pn, 0, 0);  // global_prefetch_b8
            }
        }
#endif

        float l = 0.0f;
        float w000=0,w001=0,w010=0,w011=0,w100=0,w101=0,w110=0,w111=0;
        int   base = 0;
        if (valid) {
            // align_corners=True trilinear setup (f computed AFTER clamping, as in ref)
            float ix = px * 159.0f, iy = py * 159.0f, iz = pz * 159.0f;
            int x0 = min(max((int)floorf(ix), 0), RES - 2);
            int y0 = min(max((int)floorf(iy), 0), RES - 2);
            int z0 = min(max((int)floorf(iz), 0), RES - 2);
            float fx = ix - (float)x0, fy = iy - (float)y0, fz = iz - (float)z0;
            float gx = 1.0f - fx, gy = 1.0f - fy, gz = 1.0f - fz;
            w000 = gx*gy*gz; w001 = gx*gy*fz; w010 = gx*fy*gz; w011 = gx*fy*fz;
            w100 = fx*gy*gz; w101 = fx*gy*fz; w110 = fx*fy*gz; w111 = fx*fy*fz;
            base = x0 * RES2 + y0 * RES + z0;

            const float* g = dgrid + base;
            float dens = g[0]     * w000 + g[1]        * w001
                       + g[RES]   * w010 + g[RES+1]    * w011
                       + g[RES2]  * w100 + g[RES2+1]   * w101
                       + g[RES2+RES]*w110 + g[RES2+RES+1]*w111;
            l = -0.5f * softplus_f(dens + ACT_SHIFT_F);   // log(1-alpha)
        }

        // wave32 inclusive scan of l across lanes (CDNA5 ds_permute path)
        float incl = l;
        #pragma unroll
        for (int off = 1; off < 32; off <<= 1) {
            float n = __shfl_up(incl, off, 32);
            if (lane >= off) incl += n;
        }
        float excl  = incl - l;
        float total = __shfl(incl, 31, 32);

        if (valid) {
            float T     = expf(cum + excl);   // transmittance before this sample
            float alpha = -expm1f(l);         // 1 - exp(l)
            float w     = T * alpha;

            const float* g0 = kgrid + base;
            #pragma unroll
            for (int c = 0; c < 3; ++c) {
                const float* g = g0 + c * RES3;
                float v = g[0]     * w000 + g[1]        * w001
                        + g[RES]   * w010 + g[RES+1]    * w011
                        + g[RES2]  * w100 + g[RES2+1]   * w101
                        + g[RES2+RES]*w110 + g[RES2+RES+1]*w111;
                float rgb = 1.0f / (1.0f + expf(-v));   // sigmoid
                float contrib = w * rgb;
                if      (c == 0) a0 += contrib;
                else if (c == 1) a1 += contrib;
                else             a2 += contrib;
            }
        }
        cum += total;
    }

    // butterfly reduce the per-lane partial sums
    #pragma unroll
    for (int off = 16; off > 0; off >>= 1) {
        a0 += __shfl_down(a0, off, 32);
        a1 += __shfl_down(a1, off, 32);
        a2 += __shfl_down(a2, off, 32);
    }
    if (lane == 0) {
        float bg = expf(cum);                 // alphainv_last * BKGD (BKGD = 1)
        out[3 * (size_t)ray + 0] = a0 + bg;
        out[3 * (size_t)ray + 1] = a1 + bg;
        out[3 * (size_t)ray + 2] = a2 + bg;
    }
}

// ---------------- host launcher ----------------
extern "C" void kernel_launch(void* const* d_in, const int* in_sizes, int n_in,
                              void* d_out, int out_size, void* d_ws, size_t ws_size,
                              hipStream_t stream) {
    const float* xyz    = (const float*)d_in[0];   // [M,3]
    const float* dgrid  = (const float*)d_in[1];   // [1,160,160,160]
    const float* kgrid  = (const float*)d_in[2];   // [3,160,160,160]
    const int*   ray_id = (const int*)  d_in[3];   // [M], sorted
    // d_in[4] = n_rays scalar on device; derive from out_size instead.

    const int M      = in_sizes[0] / 3;
    const int n_rays = out_size / 3;

    int* bounds = (int*)d_ws;                      // (n_rays + 1) ints

    {
        int threads = 256;
        int blocks  = (n_rays + 1 + threads - 1) / threads;
        bounds_kernel<<<blocks, threads, 0, stream>>>(ray_id, M, n_rays, bounds);
    }
    {
        int blocks = (n_rays + WPB - 1) / WPB;     // one wave32 per ray
        render_kernel<<<blocks, BLOCK, 0, stream>>>(xyz, dgrid, kgrid, bounds,
                                                    (float*)d_out, n_rays);
    }
}